// BaselineDNN_43834436223012
// MI455X (gfx1250) — compile-verified
//
#include <hip/hip_runtime.h>
#include <hip/hip_bf16.h>

#define VOCAB  400000
#define EMB    300
#define BATCH  4096
#define SEQ    200
#define REP    (2 * EMB)   // 600
#define HID    32
#define OUTD   5

typedef __attribute__((ext_vector_type(2))) float v2f;
typedef __attribute__((ext_vector_type(8))) float v8f;

// ---------------------------------------------------------------------------
// Kernel 1: embedding gather + (sum/len, max) reduction.
// One block per batch row; lane t < 300 owns embedding dim t (coalesced).
// Indices are uniform across the block -> scalar loads. Unroll x4 for MLP.
// ---------------------------------------------------------------------------
__global__ __launch_bounds__(320)
void gather_reduce_kernel(const int* __restrict__ x,
                          const int* __restrict__ lengths,
                          const float* __restrict__ emb,
                          float* __restrict__ rep) {
    const int b = blockIdx.x;
    const int t = threadIdx.x;
    if (t >= EMB) return;

    const int* xrow = x + (size_t)b * SEQ;

    float sum = 0.0f;
    float mx  = -3.402823466e38f;

    for (int s = 0; s < SEQ; s += 4) {
        const int i0 = xrow[s + 0];
        const int i1 = xrow[s + 1];
        const int i2 = xrow[s + 2];
        const int i3 = xrow[s + 3];

        // Prefetch the next group's embedding rows (global_prefetch_b8).
        if (s + 8 <= SEQ) {
            const int p0 = xrow[s + 4];
            const int p1 = xrow[s + 5];
            const int p2 = xrow[s + 6];
            const int p3 = xrow[s + 7];
            __builtin_prefetch(&emb[(size_t)p0 * EMB + t], 0, 1);
            __builtin_prefetch(&emb[(size_t)p1 * EMB + t], 0, 1);
            __builtin_prefetch(&emb[(size_t)p2 * EMB + t], 0, 1);
            __builtin_prefetch(&emb[(size_t)p3 * EMB + t], 0, 1);
        }

        const float v0 = emb[(size_t)i0 * EMB + t];
        const float v1 = emb[(size_t)i1 * EMB + t];
        const float v2 = emb[(size_t)i2 * EMB + t];
        const float v3 = emb[(size_t)i3 * EMB + t];

        sum += (v0 + v1) + (v2 + v3);
        mx = fmaxf(fmaxf(mx, v0), fmaxf(v1, fmaxf(v2, v3)));
    }

    const float inv_len = 1.0f / (float)lengths[b];
    rep[(size_t)b * REP + t]       = sum * inv_len;   // mean part (cols 0..299)
    rep[(size_t)b * REP + EMB + t] = mx;              // max part  (cols 300..599)
}

// ---------------------------------------------------------------------------
// Kernel 2: MLP head. One wave32 per 16-row tile of rep.
// Layer 1 (600x32) via V_WMMA_F32_16X16X4_F32 (f32 in/out, matches reference
// precision), two 16x16 N-tiles, 150 K-steps. Bias+ReLU on the accumulator,
// stage h through LDS, then the tiny 32->5 layer as scalar dots.
// ---------------------------------------------------------------------------
__global__ __launch_bounds__(32)
void mlp_wmma_kernel(const float* __restrict__ rep,
                     const float* __restrict__ W1,
                     const float* __restrict__ b1,
                     const float* __restrict__ W2,
                     const float* __restrict__ b2,
                     float* __restrict__ out) {
    __shared__ float h_lds[16][HID];

    const int lane     = threadIdx.x;      // 0..31
    const int half     = lane >> 4;        // 0 or 1
    const int m        = lane & 15;        // 0..15
    const int row_base = blockIdx.x * 16;

    v8f c0 = {};
    v8f c1 = {};

    const float* arow = rep + (size_t)(row_base + m) * REP;

    for (int k = 0; k < REP; k += 4) {
        const int ka = k + 2 * half;       // K index base for this lane half

        // A fragment (16x4): lane half 0 -> K=k,k+1 ; half 1 -> K=k+2,k+3
        v2f a;
        a.x = arow[ka + 0];
        a.y = arow[ka + 1];

        // B fragments (4x16): VGPR0 holds K=ka, VGPR1 holds K=ka+1, col = m (+16)
        const float* w1r0 = W1 + (size_t)(ka + 0) * HID;
        const float* w1r1 = W1 + (size_t)(ka + 1) * HID;
        v2f bf0, bf1;
        bf0.x = w1r0[m];        bf0.y = w1r1[m];         // N-tile 0 (cols 0..15)
        bf1.x = w1r0[m + 16];   bf1.y = w1r1[m + 16];    // N-tile 1 (cols 16..31)

        c0 = __builtin_amdgcn_wmma_f32_16x16x4_f32(false, a, false, bf0,
                                                   (short)0, c0, false, false);
        c1 = __builtin_amdgcn_wmma_f32_16x16x4_f32(false, a, false, bf1,
                                                   (short)0, c1, false, false);
    }

    // C/D layout: VGPR v, lanes 0-15 -> row v col m; lanes 16-31 -> row v+8 col m
    const float bias0 = b1[m];
    const float bias1 = b1[m + 16];
#pragma unroll
    for (int v = 0; v < 8; ++v) {
        const int r = v + 8 * half;
        h_lds[r][m]      = fmaxf(c0[v] + bias0, 0.0f);
        h_lds[r][m + 16] = fmaxf(c1[v] + bias1, 0.0f);
    }
    __syncthreads();

    // Layer 2: 16 rows x 5 outputs = 80 dot products over 32.
    for (int j = lane; j < 16 * OUTD; j += 32) {
        const int r = j / OUTD;
        const int o = j % OUTD;
        float acc = b2[o];
#pragma unroll
        for (int kk = 0; kk < HID; ++kk)
            acc += h_lds[r][kk] * W2[kk * OUTD + o];
        out[(size_t)(row_base + r) * OUTD + o] = acc;
    }
}

// ---------------------------------------------------------------------------
extern "C" void kernel_launch(void* const* d_in, const int* in_sizes, int n_in,
                              void* d_out, int out_size, void* d_ws, size_t ws_size,
                              hipStream_t stream) {
    const int*   x       = (const int*)d_in[0];    // [4096, 200]
    const int*   lengths = (const int*)d_in[1];    // [4096]
    const float* emb     = (const float*)d_in[2];  // [400000, 300]
    const float* W1      = (const float*)d_in[3];  // [600, 32]
    const float* b1      = (const float*)d_in[4];  // [32]
    const float* W2      = (const float*)d_in[5];  // [32, 5]
    const float* b2      = (const float*)d_in[6];  // [5]
    float*       out     = (float*)d_out;          // [4096, 5]
    float*       rep     = (float*)d_ws;           // [4096, 600] f32 scratch

    gather_reduce_kernel<<<BATCH, 320, 0, stream>>>(x, lengths, emb, rep);
    mlp_wmma_kernel<<<BATCH / 16, 32, 0, stream>>>(rep, W1, b1, W2, b2, out);
}